// GraphEncoder_27702539059394
// MI455X (gfx1250) — compile-verified
//
#include <hip/hip_runtime.h>

#define NN 100000
#define NE 1600000
#define DIM 128
#define EPSBN 1e-5f

typedef __attribute__((ext_vector_type(2))) float v2f;
typedef __attribute__((ext_vector_type(8))) float v8f;

__device__ __forceinline__ float fatomic_add(float* p, float v) {
  // hardware global_atomic_add_f32 (L2 atomic), not a CAS loop
  return unsafeAtomicAdd(p, v);
}

// ---------------------------------------------------------------------------
// h = relu(x @ W_in + b_in)   -- one wave per 16x16 tile, fp32 WMMA, K=128
// ---------------------------------------------------------------------------
__global__ void __launch_bounds__(256) k_gemm_in(
    const float* __restrict__ X, const float* __restrict__ W,
    const float* __restrict__ B, float* __restrict__ H) {
  const int row0 = blockIdx.x << 4;
  const int wave = threadIdx.x >> 5;
  const int lane = threadIdx.x & 31;
  const int m    = lane & 15;
  const int hi   = lane >> 4;    // 0 or 1
  const int kh   = hi << 1;      // K sub-offset 0 or 2
  const int col  = (wave << 4) + m;

  const float* arow = X + (size_t)(row0 + m) * DIM;
  v8f acc = {};
  for (int k = 0; k < DIM; k += 4) {
    const int kk = k + kh;
    v2f a = *(const v2f*)(arow + kk);
    v2f b;
    b.x = W[(size_t)kk * DIM + col];
    b.y = W[(size_t)(kk + 1) * DIM + col];
    acc = __builtin_amdgcn_wmma_f32_16x16x4_f32(false, a, false, b,
                                                (short)0, acc, false, false);
  }
  const float bias = B[col];
  const int crow = row0 + (hi << 3);
#pragma unroll
  for (int r = 0; r < 8; ++r) {
    float v = acc[r] + bias;
    H[(size_t)(crow + r) * DIM + col] = fmaxf(v, 0.0f);
  }
}

// ---------------------------------------------------------------------------
// zero a float range (grid-stride)
// ---------------------------------------------------------------------------
__global__ void k_zero(float* __restrict__ p, long n) {
  long i = (long)blockIdx.x * blockDim.x + threadIdx.x;
  const long stride = (long)gridDim.x * blockDim.x;
  for (; i < n; i += stride) p[i] = 0.0f;
}

// ---------------------------------------------------------------------------
// edge aggregation: one wave per edge, 32 lanes x float4 = 128 dims
// XN[col] += h[row] * ew ; DEG[col] += ew   (fp32 hardware atomics, L2-resident)
// ---------------------------------------------------------------------------
__global__ void __launch_bounds__(256) k_aggregate(
    const float* __restrict__ H, const int* __restrict__ src,
    const int* __restrict__ dst, const float* __restrict__ EW,
    float* __restrict__ XN, float* __restrict__ DEG) {
  const int e = blockIdx.x * 8 + (threadIdx.x >> 5);
  if (e >= NE) return;
  const int lane = threadIdx.x & 31;
  const int r = src[e];
  const int c = dst[e];
  const float w = EW[e];
  const float4 v = ((const float4*)(H + (size_t)r * DIM))[lane];
  float* out = XN + (size_t)c * DIM + lane * 4;
  fatomic_add(out + 0, v.x * w);
  fatomic_add(out + 1, v.y * w);
  fatomic_add(out + 2, v.z * w);
  fatomic_add(out + 3, v.w * w);
  if (lane == 0) fatomic_add(DEG + c, w);
}

// ---------------------------------------------------------------------------
// z = h @ Ws + (x_neigh/deg) @ Wn + bs + bn   (two fp32 WMMA chains per tile)
// fused column sum / sum-of-squares for BatchNorm batch statistics
// ---------------------------------------------------------------------------
__global__ void __launch_bounds__(256) k_combine(
    const float* __restrict__ H, const float* __restrict__ XN,
    const float* __restrict__ DEG,
    const float* __restrict__ WS, const float* __restrict__ WN,
    const float* __restrict__ BS, const float* __restrict__ BN_,
    float* __restrict__ Z, float* __restrict__ CS, float* __restrict__ CSS) {
  const int row0 = blockIdx.x << 4;
  const int wave = threadIdx.x >> 5;
  const int lane = threadIdx.x & 31;
  const int m    = lane & 15;
  const int hi   = lane >> 4;
  const int kh   = hi << 1;
  const int col  = (wave << 4) + m;

  const float rdeg = 1.0f / fmaxf(DEG[row0 + m], 1.0f);
  const float* hrow = H  + (size_t)(row0 + m) * DIM;
  const float* nrow = XN + (size_t)(row0 + m) * DIM;

  v8f acc = {};
  for (int k = 0; k < DIM; k += 4) {
    const int kk = k + kh;
    v2f a = *(const v2f*)(hrow + kk);
    v2f b;
    b.x = WS[(size_t)kk * DIM + col];
    b.y = WS[(size_t)(kk + 1) * DIM + col];
    acc = __builtin_amdgcn_wmma_f32_16x16x4_f32(false, a, false, b,
                                                (short)0, acc, false, false);
    v2f an = *(const v2f*)(nrow + kk);
    an.x *= rdeg; an.y *= rdeg;
    v2f bn;
    bn.x = WN[(size_t)kk * DIM + col];
    bn.y = WN[(size_t)(kk + 1) * DIM + col];
    acc = __builtin_amdgcn_wmma_f32_16x16x4_f32(false, an, false, bn,
                                                (short)0, acc, false, false);
  }
  const float bias = BS[col] + BN_[col];
  const int crow = row0 + (hi << 3);
  float s = 0.0f, ss = 0.0f;
#pragma unroll
  for (int r = 0; r < 8; ++r) {
    float v = acc[r] + bias;
    Z[(size_t)(crow + r) * DIM + col] = v;
    s  += v;
    ss += v * v;
  }
  fatomic_add(CS  + col, s);
  fatomic_add(CSS + col, ss);
}

// ---------------------------------------------------------------------------
// per-column BN finalize: scale = g*rsqrt(var+eps), shift = be - mean*scale
// ---------------------------------------------------------------------------
__global__ void k_bnfin(const float* __restrict__ CS, const float* __restrict__ CSS,
                        const float* __restrict__ G, const float* __restrict__ BE,
                        float* __restrict__ SC, float* __restrict__ SH) {
  const int c = threadIdx.x;
  const float inv_n = 1.0f / (float)NN;
  const float mean = CS[c] * inv_n;
  const float var  = CSS[c] * inv_n - mean * mean;
  const float sc   = G[c] * rsqrtf(var + EPSBN);
  SC[c] = sc;
  SH[c] = BE[c] - mean * sc;
}

// ---------------------------------------------------------------------------
// h = h + [relu](z * scale + shift)      (float4 grid-stride)
// ---------------------------------------------------------------------------
__global__ void k_update(float* __restrict__ H, const float* __restrict__ Z,
                         const float* __restrict__ SC, const float* __restrict__ SH,
                         int do_relu) {
  const long n4 = (long)NN * (DIM / 4);
  long i = (long)blockIdx.x * blockDim.x + threadIdx.x;
  const long stride = (long)gridDim.x * blockDim.x;
  for (; i < n4; i += stride) {
    const int c = (int)(i & 31) * 4;
    const float4 z  = ((const float4*)Z)[i];
    const float4 sc = *(const float4*)(SC + c);
    const float4 sh = *(const float4*)(SH + c);
    float4 h = ((float4*)H)[i];
    float a0 = z.x * sc.x + sh.x;
    float a1 = z.y * sc.y + sh.y;
    float a2 = z.z * sc.z + sh.z;
    float a3 = z.w * sc.w + sh.w;
    if (do_relu) {
      a0 = fmaxf(a0, 0.0f); a1 = fmaxf(a1, 0.0f);
      a2 = fmaxf(a2, 0.0f); a3 = fmaxf(a3, 0.0f);
    }
    h.x += a0; h.y += a1; h.z += a2; h.w += a3;
    ((float4*)H)[i] = h;
  }
}

extern "C" void kernel_launch(void* const* d_in, const int* in_sizes, int n_in,
                              void* d_out, int out_size, void* d_ws, size_t ws_size,
                              hipStream_t stream) {
  const float* x    = (const float*)d_in[0];
  const int*   ei   = (const int*)d_in[1];
  const float* ew   = (const float*)d_in[2];
  const float* W_in = (const float*)d_in[3];
  const float* b_in = (const float*)d_in[4];

  float* H  = (float*)d_out;                 // h lives in d_out, updated in place
  float* ws = (float*)d_ws;
  const size_t nf = (size_t)NN * DIM;
  float* Z   = ws;                           // [NN*DIM]
  float* XN  = ws + nf;                      // [NN*DIM]
  float* DEG = XN + nf;                      // [NN]
  float* CS  = DEG + NN;                     // [DIM]
  float* CSS = CS + DIM;                     // [DIM]
  float* SC  = CSS + DIM;                    // [DIM]
  float* SH  = SC + DIM;                     // [DIM]

  const int* src = ei;                       // edge_index[0]
  const int* dst = ei + NE;                  // edge_index[1]

  const dim3 gblk(NN / 16);                  // 6250 blocks, exact

  k_gemm_in<<<gblk, 256, 0, stream>>>(x, W_in, b_in, H);

  for (int layer = 0; layer < 3; ++layer) {
    const float* Ws = (const float*)d_in[5 + layer * 6 + 0];
    const float* bs = (const float*)d_in[5 + layer * 6 + 1];
    const float* Wn = (const float*)d_in[5 + layer * 6 + 2];
    const float* bn = (const float*)d_in[5 + layer * 6 + 3];
    const float* g  = (const float*)d_in[5 + layer * 6 + 4];
    const float* be = (const float*)d_in[5 + layer * 6 + 5];

    // zero XN + DEG + CS + CSS (contiguous in workspace)
    const long zero_n = (long)nf + NN + 2 * DIM;
    k_zero<<<2048, 256, 0, stream>>>(XN, zero_n);

    k_aggregate<<<NE / 8, 256, 0, stream>>>(H, src, dst, ew, XN, DEG);
    k_combine<<<gblk, 256, 0, stream>>>(H, XN, DEG, Ws, Wn, bs, bn, Z, CS, CSS);
    k_bnfin<<<1, DIM, 0, stream>>>(CS, CSS, g, be, SC, SH);
    k_update<<<2048, 256, 0, stream>>>(H, Z, SC, SH, layer < 2 ? 1 : 0);
  }
}